// DiagonalLinear_17918603559163
// MI455X (gfx1250) — compile-verified
//
#include <hip/hip_runtime.h>

// DiagonalLinear: out[b,s,j] = x[b,s,j] * w[j]
// B=4, S=4096, H=4096, fp32.  rows = B*S = 16384, H/4 = 1024 float4 per row.
//
// HBM-bound: ~512 MB moved -> ~22 us floor at 23.3 TB/s. Hot kernel is a
// branch-free stream, verified in disasm:
//   - 1 temporal b128 w load (scale_offset), per-thread fixed column group
//   - s_clause of 8x global_load_b128 th:TH_LOAD_NT, base + immediate
//     offsets (row stride 16384 B, max 114688 B < 2^23)
//   - v_pk_mul_f32 packed math overlapped with staged s_wait_loadcnt
//   - 8x global_store_b128 NT, same immediate-offset addressing
// wmma=0 is correct: no matrix-core mapping removes a single HBM byte.

typedef float v4f __attribute__((ext_vector_type(4)));

constexpr int ROWS_PER_BLOCK = 8;
constexpr int H4_REF         = 1024;   // H/4 for the reference shape

// ---- Hot path (reference shape) ----
__global__ __launch_bounds__(1024, 1)
void diag_scale_f32_h4096(const float* __restrict__ x,
                          const float* __restrict__ w,
                          float* __restrict__ out)
{
    const int t = threadIdx.x;                       // 0 .. 1023
    const v4f* __restrict__ xv = (const v4f*)x;
    v4f* __restrict__ ov       = (v4f*)out;

    // Fixed per-thread weight vector: one temporal 128-bit load -> 4 VGPRs.
    const v4f w4 = ((const v4f*)w)[t];

    // 8 contiguous rows per block; all offsets compile-time constants.
    const size_t base = (size_t)blockIdx.x * (size_t)(ROWS_PER_BLOCK * H4_REF)
                      + (size_t)t;

    v4f a[ROWS_PER_BLOCK];
#pragma unroll
    for (int i = 0; i < ROWS_PER_BLOCK; ++i)
        a[i] = __builtin_nontemporal_load(&xv[base + i * H4_REF]);

#pragma unroll
    for (int i = 0; i < ROWS_PER_BLOCK; ++i)
        __builtin_nontemporal_store(a[i] * w4, &ov[base + i * H4_REF]);
}

// ---- Shape-generic fallback (never launched for the reference shape).
//      2D grid: blockIdx.x covers columns, blockIdx.y strides rows, so the
//      w index is just the column -- no modulo / division expansion. ----
__global__ __launch_bounds__(256, 1)
void diag_scale_f32_generic(const float* __restrict__ x,
                            const float* __restrict__ w,
                            float* __restrict__ out,
                            int rows, int h4)
{
    const int col = blockIdx.x * blockDim.x + threadIdx.x;   // float4 column
    if (col >= h4) return;

    const v4f* __restrict__ xv = (const v4f*)x;
    v4f* __restrict__ ov       = (v4f*)out;
    const v4f w4 = ((const v4f*)w)[col];

    for (int r = blockIdx.y; r < rows; r += gridDim.y) {
        const size_t idx = (size_t)r * (size_t)h4 + (size_t)col;
        v4f xi = __builtin_nontemporal_load(&xv[idx]);
        __builtin_nontemporal_store(xi * w4, &ov[idx]);
    }
}

extern "C" void kernel_launch(void* const* d_in, const int* in_sizes, int n_in,
                              void* d_out, int out_size, void* d_ws, size_t ws_size,
                              hipStream_t stream)
{
    const float* x = (const float*)d_in[0];   // (B, S, H) fp32
    const float* w = (const float*)d_in[1];   // (H,)     fp32
    float* out     = (float*)d_out;           // (B, S, H) fp32

    const int n    = in_sizes[0];             // B*S*H = 67,108,864
    const int h    = in_sizes[1];             // 4096
    const int rows = n / h;                   // 16384

    if (h == 4 * H4_REF && rows % ROWS_PER_BLOCK == 0) {
        // Reference shape: 2048 blocks x 1024 threads (32 wave32s each).
        diag_scale_f32_h4096<<<rows / ROWS_PER_BLOCK, 1024, 0, stream>>>(x, w, out);
        return;
    }

    const int h4 = h / 4;
    dim3 grid((h4 + 255) / 256, rows > 4096 ? 4096 : (rows > 0 ? rows : 1));
    diag_scale_f32_generic<<<grid, 256, 0, stream>>>(x, w, out, rows, h4);
}